// HTv2_14559939133948
// MI455X (gfx1250) — compile-verified
//
#include <hip/hip_runtime.h>
#include <stdint.h>

// ---------------------------------------------------------------------------
// chord_block_compression(mean) + decode, MI455X (gfx1250).
//
// Roofline: 256 MiB minimum traffic, ~11 us at 23.3 TB/s; ~0.03 FLOP/byte ->
// pure HBM-bandwidth problem, WMMA not applicable (no matmul to exploit
// without inflating traffic). CDNA5 paths used instead:
//   * global_load_async_to_lds_b128 (ASYNCcnt) double-buffered input staging,
//     th:TH_LOAD_NT so the read-once input does not thrash the 192 MB L2
//   * s_wait_asynccnt ping-pong waits (async loads complete in order) --
//     barrier-free: each lane consumes exactly the LDS bytes it issued
//   * non-temporal b128 stores for the 128 MiB write-once output
//   * wave32-aware shapes: 128-thread WGs = 4 waves, float4/thread, one
//     wave-wide async b128 = 512 contiguous bytes
// ---------------------------------------------------------------------------

#define N_ 16
#define T_ 4096
#define C_ 512

#define SCAN_THREADS 256
#define PER_THREAD   (T_ / SCAN_THREADS)   // 16 timesteps per thread
#define K2_THREADS   (C_ / 4)              // 128 threads, float4 each
#define CHUNK        8                     // frames per async chunk (16 KiB)

typedef float v4f __attribute__((ext_vector_type(4)));

// One wave-wide async copy: each lane moves 16 bytes global -> LDS.
// ISA operand order: VDST = per-lane LDS byte offset, VADDR = 64-bit address.
// NT hint: input is streamed exactly once.
__device__ __forceinline__ void async_copy_b128(const float* gsrc, uint32_t lds_off) {
    asm volatile("global_load_async_to_lds_b128 %0, %1, off th:TH_LOAD_NT"
                 :: "v"(lds_off), "v"(gsrc)
                 : "memory");
}
// Wait until at most CHUNK async ops outstanding -> previous chunk complete
// (async loads complete in order within a wave).
__device__ __forceinline__ void wait_async_le_chunk() {
    asm volatile("s_wait_asynccnt 0x8" ::: "memory");
}
__device__ __forceinline__ void wait_async_zero() {
    asm volatile("s_wait_asynccnt 0x0" ::: "memory");
}

// ---------------------------------------------------------------------------
// Kernel 1: per-row scan of chord_changes -> block start frames + block count.
// blockId[t] = inclusive_cumsum(cc)[t] - cc[0]; since blockId[t]-blockId[t-1]
// == cc[t], a block boundary is exactly (t==0 || cc[t]==1).
// ---------------------------------------------------------------------------
__global__ void __launch_bounds__(SCAN_THREADS)
scan_blocks(const int* __restrict__ cc,
            int* __restrict__ bstart,   // [N_][T_] start frame of each block
            int* __restrict__ nblk)     // [N_]     number of blocks per row
{
    __shared__ int sdata[SCAN_THREADS];
    const int row = blockIdx.x;
    const int tid = threadIdx.x;
    const int* ccr = cc + (size_t)row * T_;

    int vals[PER_THREAD];
    int total = 0;
    const int t0 = tid * PER_THREAD;
#pragma unroll
    for (int j = 0; j < PER_THREAD; ++j) {
        vals[j] = ccr[t0 + j];
        total += vals[j];
    }

    sdata[tid] = total;
    __syncthreads();
    // Hillis-Steele inclusive scan over the 256 per-thread totals.
    for (int off = 1; off < SCAN_THREADS; off <<= 1) {
        int v = (tid >= off) ? sdata[tid - off] : 0;
        __syncthreads();
        sdata[tid] += v;
        __syncthreads();
    }
    const int excl = sdata[tid] - total;
    const int cc0  = ccr[0];

    int run = excl;
#pragma unroll
    for (int j = 0; j < PER_THREAD; ++j) {
        run += vals[j];                 // inclusive cumsum at t
        const int t   = t0 + j;
        const int bid = run - cc0;      // block id of frame t
        if (t == 0 || vals[j] == 1)     // exactly one writer per block
            bstart[row * T_ + bid] = t;
    }
    if (tid == SCAN_THREADS - 1)
        nblk[row] = run - cc0 + 1;      // last inclusive sum - cc0 + 1
}

// ---------------------------------------------------------------------------
// Kernel 2: one workgroup per (row, block). Stream the block's contiguous
// [len x 512] fp32 region through LDS with double-buffered async copies,
// reduce to a per-channel mean, then NT-broadcast the mean to every frame.
// Covers every element of d_out exactly once; no atomics -> deterministic.
// 32 KiB LDS / 4 waves per WG -> ~10 WGs (40 waves) resident per WGP.
// ---------------------------------------------------------------------------
__global__ void __launch_bounds__(K2_THREADS)
block_mean_broadcast(const float* __restrict__ H,
                     float* __restrict__ out,
                     const int* __restrict__ bstart,
                     const int* __restrict__ nblk)
{
    __shared__ float stage[2][CHUNK][C_];   // 32 KiB

    const int row = blockIdx.y;
    const int b   = blockIdx.x;
    const int nb  = nblk[row];
    if (b >= nb) return;                    // uniform across the workgroup

    const int start = bstart[row * T_ + b];
    const int end   = (b + 1 < nb) ? bstart[row * T_ + b + 1] : T_;
    const int len   = end - start;

    const int tid = threadIdx.x;
    const int c4  = tid * 4;                // this thread's 4 channels

    const float* src0 = H + ((size_t)row * T_ + start) * C_ + c4;
    // Low 32 bits of the generic shared address == LDS byte offset.
    const uint32_t lds0 =
        (uint32_t)(uintptr_t)(&stage[0][0][0]) + (uint32_t)(c4 * 4);

    // Issue async copies for frames [frame_off, frame_off+cnt) into buffer buf.
    auto issue = [&](int frame_off, int buf, int cnt) {
        const float* g = src0 + (size_t)frame_off * C_;
        const uint32_t l = lds0 + (uint32_t)(buf * CHUNK * C_ * 4);
        for (int f = 0; f < cnt; ++f)
            async_copy_b128(g + (size_t)f * C_, l + (uint32_t)(f * C_ * 4));
    };

    v4f sum = (v4f){0.f, 0.f, 0.f, 0.f};

    int buf    = 0;
    int done   = 0;
    int issued = (len < CHUNK) ? len : CHUNK;
    issue(0, buf, issued);                  // prologue chunk

    while (done < len) {
        const int cur = issued - done;      // frames resident in `buf`
        int nxt = len - issued;
        if (nxt > CHUNK) nxt = CHUNK;
        if (nxt > 0) issue(issued, buf ^ 1, nxt);

        // In-order completion: <=CHUNK outstanding => chunk in `buf` is done.
        if (nxt == CHUNK) wait_async_le_chunk();
        else              wait_async_zero();

        for (int f = 0; f < cur; ++f) {
            v4f v = *(const v4f*)(&stage[buf][f][c4]);  // ds_load_b128
            sum += v;
        }
        done   += cur;
        issued += nxt;
        buf ^= 1;
    }

    const float flen = (float)len;
    v4f mean;
    mean.x = sum.x / flen;
    mean.y = sum.y / flen;
    mean.z = sum.z / flen;
    mean.w = sum.w / flen;

    // Broadcast: streaming output, keep it out of L2 (working set > 192 MB).
    float* dst = out + ((size_t)row * T_ + start) * C_ + c4;
    for (int t = 0; t < len; ++t)
        __builtin_nontemporal_store(mean, (v4f*)(dst + (size_t)t * C_));
}

// ---------------------------------------------------------------------------
extern "C" void kernel_launch(void* const* d_in, const int* in_sizes, int n_in,
                              void* d_out, int out_size, void* d_ws, size_t ws_size,
                              hipStream_t stream) {
    const float* H  = (const float*)d_in[0];   // [16,4096,512] fp32
    const int*   cc = (const int*)d_in[1];     // [16,4096] int32
    float* out = (float*)d_out;                // [16,4096,512] fp32

    int* bstart = (int*)d_ws;                  // N_*T_ ints (256 KiB)
    int* nblk   = bstart + (size_t)N_ * T_;    // N_ ints

    scan_blocks<<<N_, SCAN_THREADS, 0, stream>>>(cc, bstart, nblk);

    dim3 grid2(T_, N_);                        // max T_ blocks per row
    block_mean_broadcast<<<grid2, K2_THREADS, 0, stream>>>(H, out, bstart, nblk);
}